// _CWNDefaultFirstConv_27496380629502
// MI455X (gfx1250) — compile-verified
//
#include <hip/hip_runtime.h>
#include <hip/hip_bf16.h>
#include <math.h>

typedef __attribute__((ext_vector_type(2))) float v2f;
typedef __attribute__((ext_vector_type(8))) float v8f;

#define C_DIM 128
#define LDS_STRIDE 130   // pad by 2 floats: keeps 8B alignment, spreads LDS banks

// ---------------------------------------------------------------------------
// Dense GEMM: Y[M x 128] = X[M x 128] @ W[128 x 128], fp32 via V_WMMA_F32_16X16X4_F32.
// 256 threads = 8 waves; each wave owns a 16-row strip x all 128 cols.
// W is staged transposed in LDS so each B fragment is one contiguous float2.
// ---------------------------------------------------------------------------
__global__ __launch_bounds__(256)
void gemm_xw_wmma(const float* __restrict__ X, const float* __restrict__ W,
                  float* __restrict__ Y, int M) {
    __shared__ float lds_wt[C_DIM * LDS_STRIDE];  // wt[n][k] = W[k][n]

    for (int i = threadIdx.x; i < C_DIM * C_DIM; i += 256) {
        int k = i >> 7;
        int n = i & 127;
        lds_wt[n * LDS_STRIDE + k] = W[i];
    }
    __syncthreads();

    const int lane = threadIdx.x & 31;
    const int wave = threadIdx.x >> 5;
    const int m_base = (blockIdx.x * 8 + wave) * 16;
    if (m_base >= M) return;               // wave-uniform exit: EXEC stays all-1 for WMMA

    const int half = lane >> 4;            // 0: lanes 0-15, 1: lanes 16-31
    const int l16  = lane & 15;

    // A 16x4 layout: lanes 0-15 carry (K=k,K=k+1) of row M=l16; lanes 16-31 carry (K=k+2,K=k+3)
    int arow = m_base + l16;
    if (arow >= M) arow = M - 1;           // clamp reads; stores are predicated below
    const float* xrow = X + (size_t)arow * C_DIM;

    v8f acc[8] = {};                       // 8 N-tiles of 16 cols -> full 128 cols

    #pragma unroll 4
    for (int k = 0; k < C_DIM; k += 4) {
        v2f a = *(const v2f*)(xrow + k + half * 2);
        #pragma unroll
        for (int t = 0; t < 8; ++t) {
            int n = t * 16 + l16;
            // B 4x16: VGPR0 = rows K=k (lanes 0-15) / K=k+2 (lanes 16-31), VGPR1 = K=k+1 / K=k+3
            v2f b = *(const v2f*)(&lds_wt[n * LDS_STRIDE + k + half * 2]);
            acc[t] = __builtin_amdgcn_wmma_f32_16x16x4_f32(
                false, a, false, b, (short)0, acc[t], false, false);
        }
    }

    // C/D 16x16 layout: VGPR r, lanes 0-15 -> (M=r, N), lanes 16-31 -> (M=r+8, N)
    #pragma unroll
    for (int t = 0; t < 8; ++t) {
        #pragma unroll
        for (int r = 0; r < 8; ++r) {
            int m = m_base + r + half * 8;
            if (m < M) Y[(size_t)m * C_DIM + t * 16 + l16] = acc[t][r];
        }
    }
}

// ---------------------------------------------------------------------------
// COO SpMM scatter: acc[rows[e], :] += vals[e] * Y[cols[e], :]
// One wave per edge; each lane moves a float4 (128 floats / 32 lanes).
// acc (51 MB) is L2-resident on MI455X (192 MB L2) so atomics resolve in L2.
// ---------------------------------------------------------------------------
__global__ __launch_bounds__(256)
void spmm_scatter(const float* __restrict__ vals, const int* __restrict__ rows,
                  const int* __restrict__ cols, const float* __restrict__ Y,
                  float* __restrict__ acc, int nnz) {
    int e = blockIdx.x * 8 + (threadIdx.x >> 5);
    if (e >= nnz) return;
    int lane = threadIdx.x & 31;

    float v = vals[e];
    int r = rows[e];
    int c = cols[e];

    const float4 d = ((const float4*)(Y + (size_t)c * C_DIM))[lane];
    float* dst = acc + (size_t)r * C_DIM + lane * 4;
    atomicAdd(dst + 0, v * d.x);
    atomicAdd(dst + 1, v * d.y);
    atomicAdd(dst + 2, v * d.z);
    atomicAdd(dst + 3, v * d.w);
}

__global__ __launch_bounds__(256)
void zero_f32(float* __restrict__ p, size_t n) {
    size_t i = (size_t)blockIdx.x * 256 + threadIdx.x;
    if (i < n) p[i] = 0.0f;
}

__device__ __forceinline__ float elu1(float x) {
    return x > 0.0f ? x : expm1f(x);
}

// out = elu(out) + elu(acc2), in place on d_out
__global__ __launch_bounds__(256)
void combine_elu(float* __restrict__ out, const float* __restrict__ acc2, size_t n) {
    size_t i = (size_t)blockIdx.x * 256 + threadIdx.x;
    if (i < n) out[i] = elu1(out[i]) + elu1(acc2[i]);
}

extern "C" void kernel_launch(void* const* d_in, const int* in_sizes, int n_in,
                              void* d_out, int out_size, void* d_ws, size_t ws_size,
                              hipStream_t stream) {
    const float* x1      = (const float*)d_in[0];
    const float* x2      = (const float*)d_in[1];
    const float* W1      = (const float*)d_in[2];
    const float* W2      = (const float*)d_in[3];
    const float* vals11  = (const float*)d_in[4];
    const float* vals21  = (const float*)d_in[5];
    const int*   rows11  = (const int*)d_in[6];
    const int*   cols11  = (const int*)d_in[7];
    const int*   rows21  = (const int*)d_in[8];
    const int*   cols21  = (const int*)d_in[9];

    const int M1  = in_sizes[0] / C_DIM;   // 100000 (n_r)
    const int M2  = in_sizes[1] / C_DIM;   // 200000
    const int nnz = in_sizes[4];           // 1.6M

    float* out  = (float*)d_out;                       // acc for branch 1
    float* y1   = (float*)d_ws;                        // x1 @ W1
    float* y2   = y1 + (size_t)M1 * C_DIM;             // x2 @ W2
    float* acc2 = y2 + (size_t)M2 * C_DIM;             // acc for branch 2

    const size_t n_out = (size_t)M1 * C_DIM;

    // 1) zero accumulators
    {
        int g = (int)((n_out + 255) / 256);
        zero_f32<<<g, 256, 0, stream>>>(out, n_out);
        zero_f32<<<g, 256, 0, stream>>>(acc2, n_out);
    }

    // 2) dense GEMMs (WMMA fp32)
    {
        int g1 = (M1 + 127) / 128;
        int g2 = (M2 + 127) / 128;
        gemm_xw_wmma<<<g1, 256, 0, stream>>>(x1, W1, y1, M1);
        gemm_xw_wmma<<<g2, 256, 0, stream>>>(x2, W2, y2, M2);
    }

    // 3) sparse scatter-adds
    {
        int g = (nnz + 7) / 8;
        spmm_scatter<<<g, 256, 0, stream>>>(vals11, rows11, cols11, y1, out,  nnz);
        spmm_scatter<<<g, 256, 0, stream>>>(vals21, rows21, cols21, y2, acc2, nnz);
    }

    // 4) out = elu(out) + elu(acc2)
    {
        int g = (int)((n_out + 255) / 256);
        combine_elu<<<g, 256, 0, stream>>>(out, acc2, n_out);
    }
}